// GTLayer_84043920048505
// MI455X (gfx1250) — compile-verified
//
#include <hip/hip_runtime.h>
#include <hip/hip_bf16.h>

typedef float v2f __attribute__((ext_vector_type(2)));
typedef float v8f __attribute__((ext_vector_type(8)));

#define LATENT 128
#define HEADS  8
#define DH     16
#define TILES_PER_BLOCK 16   // 8 waves x 2 M-tiles each

// ---------------------------------------------------------------------------
// Kernel 1: fused Q/K/V projection GEMM, native fp32 WMMA (16x16x4).
// - W staged in LDS once per block, pair-interleaved so each 4x16 B-fragment
//   half-wave read is a single conflict-free ds b64.
// - Each wave computes two 16-row M-tiles (32x128 strip): every B fragment
//   feeds two v_wmma ops.
// - matId is uniform per block; ragged tail handled by clamped A loads and
//   predicated stores only -> EXEC all-ones at every WMMA.
// ---------------------------------------------------------------------------
__global__ void __launch_bounds__(256)
qkv_gemm_wmma(const float* __restrict__ embeds,
              const float* __restrict__ Wq,
              const float* __restrict__ Wk,
              const float* __restrict__ Wv,
              float* __restrict__ Qout,
              float* __restrict__ Kout,
              float* __restrict__ Vout,
              int n_nodes) {
    __shared__ float ldsW[LATENT * LATENT];   // 64 KB, pair-interleaved layout

    const int tid    = threadIdx.x;
    const int lane   = tid & 31;
    const int wave   = tid >> 5;
    const int mTiles = (n_nodes + 15) >> 4;

    const int blocksPerMat = gridDim.x / 3;
    const int matId  = blockIdx.x / blocksPerMat;          // block-uniform
    const int blkTile = (blockIdx.x % blocksPerMat) * TILES_PER_BLOCK;

    const float* __restrict__ W   = (matId == 0) ? Wq : (matId == 1) ? Wk : Wv;
    float*       __restrict__ Out = (matId == 0) ? Qout : (matId == 1) ? Kout : Vout;

    // ---- cooperative LDS fill: lds[p*256 + col*2 + r] = W[2p + r][col] ----
    // chunk = (pair-row p, 4-col group c): b128 loads of both rows, two b128
    // interleaved stores.
#pragma unroll
    for (int i = 0; i < 8; ++i) {
        const int chunk = i * 256 + tid;          // 2048 chunks total
        const int p = chunk >> 5;                 // pair row (0..63)
        const int c = (chunk & 31) << 2;          // col group (0,4,..,124)
        const float4 g0 = *(const float4*)(W + (size_t)(2 * p)     * LATENT + c);
        const float4 g1 = *(const float4*)(W + (size_t)(2 * p + 1) * LATENT + c);
        float4* d = (float4*)(ldsW + p * 256 + c * 2);
        d[0] = make_float4(g0.x, g1.x, g0.y, g1.y);
        d[1] = make_float4(g0.z, g1.z, g0.w, g1.w);
    }
    __syncthreads();

    const int half = lane >> 4;                   // 0: lanes 0-15, 1: lanes 16-31
    const int l16  = lane & 15;

    const int t0 = blkTile + wave * 2 + 0;        // two M-tiles per wave
    const int t1 = blkTile + wave * 2 + 1;
    const int m0 = t0 << 4;
    const int m1 = t1 << 4;

    // Clamp A-load rows for ragged tail (stores predicated below).
    int aR0 = m0 + l16; if (aR0 >= n_nodes) aR0 = n_nodes - 1;
    int aR1 = m1 + l16; if (aR1 >= n_nodes) aR1 = n_nodes - 1;
    const float* __restrict__ aBase0 = embeds + (size_t)aR0 * LATENT;
    const float* __restrict__ aBase1 = embeds + (size_t)aR1 * LATENT;

    v8f acc0[8], acc1[8];
#pragma unroll
    for (int n = 0; n < 8; ++n) {
        acc0[n] = (v8f){0.f,0.f,0.f,0.f,0.f,0.f,0.f,0.f};
        acc1[n] = (v8f){0.f,0.f,0.f,0.f,0.f,0.f,0.f,0.f};
    }

    for (int k0 = 0; k0 < LATENT; k0 += 4) {
        // A fragment (16x4): lanes 0-15 carry K=k0,k0+1; lanes 16-31 K=k0+2,k0+3
        const int ka = k0 + (half << 1);          // even
        const v2f a0 = *(const v2f*)(aBase0 + ka);
        const v2f a1 = *(const v2f*)(aBase1 + ka);
        const float* __restrict__ bBase = ldsW + (ka >> 1) * 256;
#pragma unroll
        for (int n = 0; n < 8; ++n) {
            // B fragment (4x16): rows ka,ka+1 for this half-wave, N across lanes;
            // pair-interleaved layout -> one b64 LDS read.
            const v2f b = *(const v2f*)(bBase + (((n << 4) + l16) << 1));
            acc0[n] = __builtin_amdgcn_wmma_f32_16x16x4_f32(
                false, a0, false, b, (short)0, acc0[n], false, false);
            acc1[n] = __builtin_amdgcn_wmma_f32_16x16x4_f32(
                false, a1, false, b, (short)0, acc1[n], false, false);
        }
    }

    // C/D layout: VGPR i -> M = i (lanes 0-15) / i+8 (lanes 16-31), N = lane%16
#pragma unroll
    for (int n = 0; n < 8; ++n) {
        const int col = (n << 4) + l16;
#pragma unroll
        for (int i = 0; i < 8; ++i) {
            const int r0 = m0 + i + (half << 3);
            const int r1 = m1 + i + (half << 3);
            if (r0 < n_nodes) Out[(size_t)r0 * LATENT + col] = acc0[n][i];
            if (r1 < n_nodes) Out[(size_t)r1 * LATENT + col] = acc1[n][i];
        }
    }
}

// ---------------------------------------------------------------------------
// Kernel 2: per-(edge,head) attention logits -> exp -> segment-sum normalizer.
// Q/K (51 MB total) are L2-resident on the 192 MB L2; gathers are 64B reads.
// ---------------------------------------------------------------------------
__global__ void edge_att(const float* __restrict__ Q,
                         const float* __restrict__ K,
                         const int* __restrict__ rows,
                         const int* __restrict__ cols,
                         float* __restrict__ expAtt,   // [E, HEADS] (d_out att slice)
                         float* __restrict__ norm,     // [N, HEADS]
                         int nEH) {
    const int gid = blockIdx.x * blockDim.x + threadIdx.x;
    if (gid >= nEH) return;
    const int e = gid >> 3;
    const int h = gid & 7;
    const int r = rows[e];
    const int c = cols[e];

    const float4* __restrict__ q = (const float4*)(Q + (size_t)r * LATENT + h * DH);
    const float4* __restrict__ k = (const float4*)(K + (size_t)c * LATENT + h * DH);
    float dot = 0.f;
#pragma unroll
    for (int i = 0; i < 4; ++i) {
        const float4 qa = q[i];
        const float4 ka = k[i];
        dot = fmaf(qa.x, ka.x, dot);
        dot = fmaf(qa.y, ka.y, dot);
        dot = fmaf(qa.z, ka.z, dot);
        dot = fmaf(qa.w, ka.w, dot);
    }
    dot = fminf(10.f, fmaxf(-10.f, dot));
    const float ex = expf(dot);
    expAtt[gid] = ex;
    atomicAdd(&norm[r * HEADS + h], ex);          // global_atomic_add_f32, no-return
}

// ---------------------------------------------------------------------------
// Kernel 3: normalize att in place and scatter att * V[col] into resEmbeds[row].
// ---------------------------------------------------------------------------
__global__ void edge_aggregate(const float* __restrict__ V,
                               const int* __restrict__ rows,
                               const int* __restrict__ cols,
                               float* __restrict__ att,        // in: expAtt, out: normalized
                               const float* __restrict__ norm, // [N, HEADS]
                               float* __restrict__ outEmb,     // [N, LATENT], zero-initialized
                               int nEH) {
    const int gid = blockIdx.x * blockDim.x + threadIdx.x;
    if (gid >= nEH) return;
    const int e = gid >> 3;
    const int h = gid & 7;
    const int r = rows[e];
    const int c = cols[e];

    const float a = att[gid] / (norm[r * HEADS + h] + 1e-8f);
    att[gid] = a;

    const float4* __restrict__ v = (const float4*)(V + (size_t)c * LATENT + h * DH);
    float* __restrict__ dst = outEmb + (size_t)r * LATENT + h * DH;
#pragma unroll
    for (int i = 0; i < 4; ++i) {
        const float4 vv = v[i];
        atomicAdd(dst + i * 4 + 0, a * vv.x);
        atomicAdd(dst + i * 4 + 1, a * vv.y);
        atomicAdd(dst + i * 4 + 2, a * vv.z);
        atomicAdd(dst + i * 4 + 3, a * vv.w);
    }
}

// ---------------------------------------------------------------------------
// Launch: d_in order = embeds, rows, cols, Wq, Wk, Wv, n_nodes
// d_out = resEmbeds [N*128] ++ att [E*8]
// d_ws  = Q [N*128] | K [N*128] | V [N*128] | norm [N*8]   (~78.4 MB)
// ---------------------------------------------------------------------------
extern "C" void kernel_launch(void* const* d_in, const int* in_sizes, int n_in,
                              void* d_out, int out_size, void* d_ws, size_t ws_size,
                              hipStream_t stream) {
    const float* embeds = (const float*)d_in[0];
    const int*   rows   = (const int*)d_in[1];
    const int*   cols   = (const int*)d_in[2];
    const float* Wq     = (const float*)d_in[3];
    const float* Wk     = (const float*)d_in[4];
    const float* Wv     = (const float*)d_in[5];

    const int N = in_sizes[0] / LATENT;   // 50000
    const int E = in_sizes[1];            // 800000

    float* Qw   = (float*)d_ws;
    float* Kw   = Qw + (size_t)N * LATENT;
    float* Vw   = Kw + (size_t)N * LATENT;
    float* norm = Vw + (size_t)N * LATENT;

    float* outEmb = (float*)d_out;                       // [N, LATENT]
    float* attOut = outEmb + (size_t)N * LATENT;         // [E, HEADS]

    hipMemsetAsync(outEmb, 0, (size_t)N * LATENT * sizeof(float), stream);
    hipMemsetAsync(norm,   0, (size_t)N * HEADS  * sizeof(float), stream);

    // GEMM: 3 matrices, 16 M-tiles (256 rows) per block, 256 threads = 8 waves
    const int mTiles = (N + 15) / 16;
    const int blocksPerMat = (mTiles + TILES_PER_BLOCK - 1) / TILES_PER_BLOCK;
    qkv_gemm_wmma<<<3 * blocksPerMat, 256, 0, stream>>>(
        embeds, Wq, Wk, Wv, Qw, Kw, Vw, N);

    const int nEH = E * HEADS;
    const int blk = 256;
    edge_att<<<(nEH + blk - 1) / blk, blk, 0, stream>>>(Qw, Kw, rows, cols, attOut, norm, nEH);
    edge_aggregate<<<(nEH + blk - 1) / blk, blk, 0, stream>>>(Vw, rows, cols, attOut, norm, outEmb, nEH);
}